// Net_29643864277323
// MI455X (gfx1250) — compile-verified
//
#include <hip/hip_runtime.h>
#include <math.h>

#define BGRAPHS 512
#define NNODE   100
#define NT      (BGRAPHS * NNODE)   // 51200
#define EPG     1600
#define NEDGE   (BGRAPHS * EPG)     // 819200
#define D       128
#define K1      80
#define K2      64

typedef float v2f __attribute__((ext_vector_type(2)));
typedef float v8f __attribute__((ext_vector_type(8)));

// ---------------------------------------------------------------------------
// Kernel 1: embedding gather  h[n][:] = emb[x_idx[n]][:]
// one thread per float4 chunk: 51200 * 32 threads
// ---------------------------------------------------------------------------
__global__ void gather_kernel(const int* __restrict__ x_idx,
                              const float* __restrict__ emb,
                              float* __restrict__ h) {
    int t = blockIdx.x * blockDim.x + threadIdx.x;
    int n = t >> 5;
    int c = (t & 31) << 2;
    if (n >= NT) return;
    int v = x_idx[n];
    const float4 val = *(const float4*)(emb + v * D + c);
    *(float4*)(h + n * D + c) = val;
}

// ---------------------------------------------------------------------------
// Kernel 2: edge scatter   agg[dst] += h[src];  deg[dst] += 1
// 32 threads per edge, 4 floats per thread, hardware f32 atomics (L2 resident)
// ---------------------------------------------------------------------------
__global__ void scatter_kernel(const float* __restrict__ h,
                               const int* __restrict__ src,
                               const int* __restrict__ dst,
                               float* __restrict__ agg,
                               float* __restrict__ deg) {
    int t = blockIdx.x * blockDim.x + threadIdx.x;
    int e = t >> 5;
    int c = (t & 31) << 2;
    if (e >= NEDGE) return;
    int s = src[e];
    int d = dst[e];
    const float4 v = *(const float4*)(h + s * D + c);
    float* out = agg + d * D + c;
    atomicAdd(out + 0, v.x);
    atomicAdd(out + 1, v.y);
    atomicAdd(out + 2, v.z);
    atomicAdd(out + 3, v.w);
    if ((t & 31) == 0) atomicAdd(deg + d, 1.0f);
}

// ---------------------------------------------------------------------------
// Kernel 3: fused SAGEConv GEMM via V_WMMA_F32_16X16X4_F32
//   hout = relu( (agg/max(deg,1)) @ Wl + bl + h @ Wr )
// block = 256 threads = 8 waves; each block: 16 rows x 128 cols
// each wave: one 16x16 tile, K-loop over 128 in steps of 4, two matrices
// fused into one accumulator.
// ---------------------------------------------------------------------------
__global__ void conv_gemm_kernel(const float* __restrict__ h,
                                 const float* __restrict__ agg,
                                 const float* __restrict__ deg,
                                 const float* __restrict__ Wl,
                                 const float* __restrict__ bl,
                                 const float* __restrict__ Wr,
                                 float* __restrict__ hout) {
    // padded stride: 132 words -> lanes with row stride hit distinct banks
    __shared__ float sH[16][132];   // raw features (A for Wr)
    __shared__ float sM[16][132];   // mean features (A for Wl)

    const int tid  = threadIdx.x;
    const int row0 = blockIdx.x * 16;

    for (int i = tid; i < 16 * D; i += 256) {
        int r = i >> 7;
        int c = i & (D - 1);
        int n = row0 + r;
        float hv = h[n * D + c];
        float rd = 1.0f / fmaxf(deg[n], 1.0f);
        sH[r][c] = hv;
        sM[r][c] = agg[n * D + c] * rd;
    }
    __syncthreads();

    const int wave  = tid >> 5;           // 0..7 -> 16-column block
    const int lane  = tid & 31;
    const int r     = lane & 15;          // A row within tile
    const int khalf = (lane >> 4) << 1;   // 0 for lanes 0-15, 2 for 16-31
    const int col   = wave * 16 + (lane & 15);

    v8f acc = {};
    for (int k = 0; k < D; k += 4) {
        v2f a, b;
        // h @ Wr
        a.x = sH[r][k + khalf];
        a.y = sH[r][k + khalf + 1];
        b.x = Wr[(k + khalf) * D + col];
        b.y = Wr[(k + khalf + 1) * D + col];
        acc = __builtin_amdgcn_wmma_f32_16x16x4_f32(
            false, a, false, b, (short)0, acc, false, false);
        // mean @ Wl
        a.x = sM[r][k + khalf];
        a.y = sM[r][k + khalf + 1];
        b.x = Wl[(k + khalf) * D + col];
        b.y = Wl[(k + khalf + 1) * D + col];
        acc = __builtin_amdgcn_wmma_f32_16x16x4_f32(
            false, a, false, b, (short)0, acc, false, false);
    }

    // C/D layout: VGPR v -> row v (lanes 0-15) or row v+8 (lanes 16-31),
    // column = 16*wave + (lane & 15)
    const float bias  = bl[col];
    const int   rbase = row0 + ((lane >> 4) << 3);
    #pragma unroll
    for (int v = 0; v < 8; ++v) {
        float val = acc[v] + bias;
        hout[(rbase + v) * D + col] = fmaxf(val, 0.0f);
    }
}

// ---------------------------------------------------------------------------
// Kernel 4: per-graph TopK pooling x2 + readouts + MLP + sigmoid
// one block (128 threads = 4 waves) per graph; graph tile lives in LDS.
// TopK membership via rank counting (readout is permutation invariant).
// ---------------------------------------------------------------------------
__global__ void pool_mlp_kernel(const float* __restrict__ h,
                                const float* __restrict__ pw1,
                                const float* __restrict__ pw2,
                                const float* __restrict__ W1,
                                const float* __restrict__ b1,
                                const float* __restrict__ W3,
                                const float* __restrict__ b3,
                                float* __restrict__ out) {
    __shared__ float sh[NNODE * D];     // 51200 B
    __shared__ float spw1[D], spw2[D];
    __shared__ float s1[NNODE], s2[NNODE], dot2s[NNODE];
    __shared__ int   keep1[NNODE], keep2[NNODE];
    __shared__ float xv[2 * D];
    __shared__ float yv[64];

    const int g    = blockIdx.x;
    const int tid  = threadIdx.x;
    const int lane = tid & 31;
    const int wave = tid >> 5;
    const float* hg = h + g * NNODE * D;

    // stage graph tile + pooling weights
    for (int i = tid; i < NNODE * D / 4; i += blockDim.x) {
        *(float4*)(sh + i * 4) = *(const float4*)(hg + i * 4);
    }
    if (tid < D) { spw1[tid] = pw1[tid]; spw2[tid] = pw2[tid]; }
    __syncthreads();

    // norms (redundant per thread; 256 FLOPs)
    float n1 = 0.0f, n2 = 0.0f;
    for (int i = 0; i < D; ++i) { n1 += spw1[i] * spw1[i]; n2 += spw2[i] * spw2[i]; }
    const float inv1 = 1.0f / sqrtf(n1);
    const float inv2 = 1.0f / sqrtf(n2);

    // per-node dots: one wave per node (strided), wave-level reduction
    for (int n = wave; n < NNODE; n += 4) {
        float a1 = 0.0f, a2 = 0.0f;
        for (int j = lane; j < D; j += 32) {
            float hv = sh[n * D + j];
            a1 += hv * spw1[j];
            a2 += hv * spw2[j];
        }
        for (int off = 16; off > 0; off >>= 1) {
            a1 += __shfl_xor(a1, off, 32);
            a2 += __shfl_xor(a2, off, 32);
        }
        if (lane == 0) {
            s1[n]    = tanhf(a1 * inv1);
            dot2s[n] = a2;
        }
    }
    __syncthreads();

    // pooling 1: rank by pairwise count, keep top K1
    if (tid < NNODE) {
        float sn = s1[tid];
        int cnt = 0;
        for (int m = 0; m < NNODE; ++m) {
            float sm = s1[m];
            cnt += (sm > sn) || (sm == sn && m < tid);
        }
        keep1[tid] = (cnt < K1);
    }
    __syncthreads();

    // pooling 2 scores: tanh( s1 * (h.pw2) / ||pw2|| ) on kept nodes
    if (tid < NNODE) {
        s2[tid] = keep1[tid] ? tanhf(s1[tid] * dot2s[tid] * inv2) : -2.0f;
    }
    __syncthreads();
    if (tid < NNODE) {
        int k2 = 0;
        if (keep1[tid]) {
            float sn = s2[tid];
            int cnt = 0;
            for (int m = 0; m < NNODE; ++m) {
                if (!keep1[m]) continue;
                float sm = s2[m];
                cnt += (sm > sn) || (sm == sn && m < tid);
            }
            k2 = (cnt < K2);
        }
        keep2[tid] = k2;
    }
    __syncthreads();

    // readouts: x = (max||mean)(hp1) + (max||mean)(hp2)
    if (tid < D) {
        const int d = tid;
        float mx1 = -INFINITY, sm1 = 0.0f;
        float mx2 = -INFINITY, sm2 = 0.0f;
        for (int n = 0; n < NNODE; ++n) {
            if (keep1[n]) {
                float v = sh[n * D + d] * s1[n];
                mx1 = fmaxf(mx1, v);
                sm1 += v;
                if (keep2[n]) {
                    float w = v * s2[n];
                    mx2 = fmaxf(mx2, w);
                    sm2 += w;
                }
            }
        }
        xv[d]     = mx1 + mx2;
        xv[D + d] = sm1 * (1.0f / (float)K1) + sm2 * (1.0f / (float)K2);
    }
    __syncthreads();

    // MLP: 256 -> 64 (relu) -> 1 (sigmoid)
    if (tid < 64) {
        float acc = b1[tid];
        for (int i = 0; i < 2 * D; ++i) acc += xv[i] * W1[i * 64 + tid];
        yv[tid] = fmaxf(acc, 0.0f);
    }
    __syncthreads();
    if (tid == 0) {
        float acc = b3[0];
        for (int j = 0; j < 64; ++j) acc += yv[j] * W3[j];
        out[g] = 1.0f / (1.0f + expf(-acc));
    }
}

// ---------------------------------------------------------------------------
extern "C" void kernel_launch(void* const* d_in, const int* in_sizes, int n_in,
                              void* d_out, int out_size, void* d_ws, size_t ws_size,
                              hipStream_t stream) {
    const int*   x_idx = (const int*)d_in[0];
    const int*   edges = (const int*)d_in[1];   // [2, NEDGE]: src then dst
    // d_in[2] = batch (unused: graphs are contiguous N-blocks)
    const float* emb = (const float*)d_in[3];
    const float* Wl  = (const float*)d_in[4];
    const float* bl  = (const float*)d_in[5];
    const float* Wr  = (const float*)d_in[6];
    const float* pw1 = (const float*)d_in[7];
    const float* pw2 = (const float*)d_in[8];
    const float* W1  = (const float*)d_in[9];
    const float* b1  = (const float*)d_in[10];
    const float* W3  = (const float*)d_in[11];
    const float* b3  = (const float*)d_in[12];

    float* ws   = (float*)d_ws;
    float* h    = ws;                    // NT*D
    float* agg  = h + (size_t)NT * D;    // NT*D
    float* deg  = agg + (size_t)NT * D;  // NT
    float* hout = deg + NT;              // NT*D

    // zero accumulators (agg and deg are contiguous)
    hipMemsetAsync(agg, 0, ((size_t)NT * D + NT) * sizeof(float), stream);

    gather_kernel<<<(NT * 32) / 256, 256, 0, stream>>>(x_idx, emb, h);

    scatter_kernel<<<(NEDGE * 32) / 256, 256, 0, stream>>>(
        h, edges, edges + NEDGE, agg, deg);

    conv_gemm_kernel<<<NT / 16, 256, 0, stream>>>(
        h, agg, deg, Wl, bl, Wr, hout);

    pool_mlp_kernel<<<BGRAPHS, 128, 0, stream>>>(
        hout, pw1, pw2, W1, b1, W3, b3, (float*)d_out);
}